// GridSpatialIntegral_20332375179585
// MI455X (gfx1250) — compile-verified
//
#include <hip/hip_runtime.h>

// GridSpatialIntegral on gfx1250 (MI455X).
// cumsum along width (ch0) and height (ch1) of [32,2,512,512] f32.
// Memory-bound (128 MiB total -> ~5.5us @ 23.3 TB/s). The scan is realized as
// blocked prefix-sum via V_WMMA_F32_16X16X4_F32 (exact fp32 matmul):
//   ch0 tile: D = X(16x16) * U(upper-tri ones), chained as 4 K=4 WMMAs
//   ch1 tile: D = L(lower-tri ones) * X(16x16)
// Inter-tile carries are injected through the WMMA accumulator C and
// propagated with one cross-lane shuffle (ds_bpermute) per D register.

typedef float v2f __attribute__((ext_vector_type(2)));
typedef float v8f __attribute__((ext_vector_type(8)));

#define IMG_W 512
#define NBATCH 32
// waves per channel: (32*512)/16 = 1024
#define WAVES_PER_CH 1024

__global__ __launch_bounds__(256)
void GridSpatialIntegral_20332375179585_kernel(const float* __restrict__ in,
                                               float* __restrict__ out) {
  const int wave = (int)((blockIdx.x * blockDim.x + threadIdx.x) >> 5);
  const int lane = (int)(threadIdx.x & 31);
  const int hi   = lane >> 4;   // half-wave: 0 = lanes 0-15, 1 = lanes 16-31
  const int ln   = lane & 15;

  if (wave < WAVES_PER_CH) {
    // ---------------- channel 0: cumsum along width (contiguous axis) -----
    // Wave handles 16 consecutive image rows. All 16 rows share one batch b
    // since 16 | 512.
    const int R0 = wave * 16;              // global x-row index of row M=0
    const int b  = R0 >> 9;                // / 512
    const int i0 = R0 & (IMG_W - 1);
    const size_t planeOff = ((size_t)(b * 2) * IMG_W + i0) * IMG_W;
    const float* src = in  + planeOff;
    float*       dst = out + planeOff;
    // A-matrix feed: lane's row is M = ln (both halves), K offset 2*hi baked in.
    const float* arow = src + (size_t)ln * IMG_W + 2 * hi;

    v8f carry = {0.f, 0.f, 0.f, 0.f, 0.f, 0.f, 0.f, 0.f};

    for (int t = 0; t < IMG_W / 16; ++t) {
      const int col0 = t * 16;
      // Load 16x16 tile in A layout: per chunk c, float2 = {K=4c+2hi, K=4c+1+2hi}
      v2f a[4];
#pragma unroll
      for (int c = 0; c < 4; ++c)
        a[c] = *(const v2f*)(arow + col0 + 4 * c);

      if (t + 1 < IMG_W / 16)
        __builtin_prefetch(arow + col0 + 16, 0, 3);   // WGP-scope global_prefetch_b8

      v8f d = carry;  // carry enters as the accumulator (rank-1 row offset)
#pragma unroll
      for (int c = 0; c < 4; ++c) {
        const int kb = 4 * c + 2 * hi;
        // B = U chunk: B[kb][n] = (kb <= n), n = ln
        v2f bm;
        bm.x = (kb     <= ln) ? 1.0f : 0.0f;
        bm.y = (kb + 1 <= ln) ? 1.0f : 0.0f;
        d = __builtin_amdgcn_wmma_f32_16x16x4_f32(false, a[c], false, bm,
                                                  (short)0, d, false, false);
      }

      // Store: D[v] -> row (v + 8*hi), cols col0 + ln  (64B contiguous / half)
#pragma unroll
      for (int v = 0; v < 8; ++v)
        dst[(size_t)(v + 8 * hi) * IMG_W + col0 + ln] = d[v];

      // New carry per row = last column (N=15): lane 15 (rows 0-7),
      // lane 31 (rows 8-15); broadcast within each half-wave.
#pragma unroll
      for (int v = 0; v < 8; ++v)
        carry[v] = __shfl(d[v], (lane & 16) | 15, 32);
    }
  } else if (wave < 2 * WAVES_PER_CH) {
    // ---------------- channel 1: cumsum along height ----------------------
    // Wave handles 16 consecutive columns of one batch's y-plane.
    const int wq = wave - WAVES_PER_CH;
    const int b  = wq >> 5;                // 32 column-groups per batch
    const int c0 = (wq & 31) * 16;
    const size_t planeOff = (size_t)(b * 2 + 1) * IMG_W * IMG_W;
    const float* src = in  + planeOff;
    float*       dst = out + planeOff;

    float carry1 = 0.0f;   // per-column carry, lane ln = column c0+ln (both halves)

    for (int t = 0; t < IMG_W / 16; ++t) {
      const int row0 = t * 16;
      // B = X chunk in B layout: K = j + 2*hi (tile row), N = ln (column).
      v2f bm[4];
#pragma unroll
      for (int c = 0; c < 4; ++c) {
        const int kb = 4 * c + 2 * hi;
        bm[c].x = src[(size_t)(row0 + kb)     * IMG_W + c0 + ln];
        bm[c].y = src[(size_t)(row0 + kb + 1) * IMG_W + c0 + ln];
      }

      if (t + 1 < IMG_W / 16)
        __builtin_prefetch(src + (size_t)(row0 + 16) * IMG_W + c0 + ln, 0, 3);

      v8f d;
#pragma unroll
      for (int v = 0; v < 8; ++v) d[v] = carry1;  // column carry into accumulator

#pragma unroll
      for (int c = 0; c < 4; ++c) {
        const int kb = 4 * c + 2 * hi;
        // A = L chunk: A[m][kb] = (kb <= m), m = ln
        v2f am;
        am.x = (kb     <= ln) ? 1.0f : 0.0f;
        am.y = (kb + 1 <= ln) ? 1.0f : 0.0f;
        d = __builtin_amdgcn_wmma_f32_16x16x4_f32(false, am, false, bm[c],
                                                  (short)0, d, false, false);
      }

#pragma unroll
      for (int v = 0; v < 8; ++v)
        dst[(size_t)(row0 + v + 8 * hi) * IMG_W + c0 + ln] = d[v];

      // New carry = bottom row (M=15) = D[7] in the high half, per column.
      carry1 = __shfl(d[7], 16 + ln, 32);
    }
  }
}

extern "C" void kernel_launch(void* const* d_in, const int* in_sizes, int n_in,
                              void* d_out, int out_size, void* d_ws, size_t ws_size,
                              hipStream_t stream) {
  (void)in_sizes; (void)n_in; (void)out_size; (void)d_ws; (void)ws_size;
  const float* in = (const float*)d_in[0];
  float* out = (float*)d_out;
  // 2048 waves total (1024 per channel) -> 65536 threads -> 256 blocks x 256.
  const int threads = 256;
  const int blocks  = (2 * WAVES_PER_CH * 32) / threads;
  GridSpatialIntegral_20332375179585_kernel<<<blocks, threads, 0, stream>>>(in, out);
}